// GNN_57303453663856
// MI455X (gfx1250) — compile-verified
//
#include <hip/hip_runtime.h>
#include <hip/hip_bf16.h>
#include <math.h>
#include <float.h>

// Problem constants (match reference)
#define NN   50000      // nodes
#define NE   800000     // edges
#define HIDF 128        // feature dim (NODE_DIM == HID == 128)

typedef __attribute__((ext_vector_type(2))) float v2f;
typedef __attribute__((ext_vector_type(8))) float v8f;

// ---------------------------------------------------------------- utilities

__global__ __launch_bounds__(256) void GNN_zero_f32(float* __restrict__ p, long n) {
    long i = (long)blockIdx.x * blockDim.x + threadIdx.x;
    long stride = (long)gridDim.x * blockDim.x;
    for (; i < n; i += stride) p[i] = 0.0f;
}

// h = nan_to_num(x): NaN->0, +inf->FLT_MAX, -inf->-FLT_MAX
__global__ __launch_bounds__(256) void GNN_nan_to_num(const float* __restrict__ x,
                                                      float* __restrict__ h, long n) {
    long i = (long)blockIdx.x * blockDim.x + threadIdx.x;
    if (i >= n) return;
    float v = x[i];
    if (isnan(v)) v = 0.0f;
    else if (isinf(v)) v = (v > 0.0f) ? FLT_MAX : -FLT_MAX;
    h[i] = v;
}

// deg[dst[e]] += 1
__global__ __launch_bounds__(256) void GNN_degree(const int* __restrict__ dst,
                                                  float* __restrict__ deg) {
    int e = blockIdx.x * blockDim.x + threadIdx.x;
    if (e >= NE) return;
    atomicAdd(&deg[dst[e]], 1.0f);
}

// in-place: norm[i] = deg>0 ? rsqrt(max(deg,1)) : 0
__global__ __launch_bounds__(256) void GNN_norm(float* __restrict__ dn) {
    int i = blockIdx.x * blockDim.x + threadIdx.x;
    if (i >= NN) return;
    float d = dn[i];
    dn[i] = (d > 0.0f) ? rsqrtf(fmaxf(d, 1.0f)) : 0.0f;
}

// out[i, :] = in[i, :] * norm[i]  (float4 vectorized; 32 threads per node)
__global__ __launch_bounds__(256) void GNN_scale(const float* __restrict__ in,
                                                 const float* __restrict__ norm,
                                                 float* __restrict__ out) {
    int t = blockIdx.x * blockDim.x + threadIdx.x;      // 0 .. NN*32-1
    int node = t >> 5;
    int g = t & 31;
    if (node >= NN) return;
    float s = norm[node];
    const float4* ip = (const float4*)(in + (size_t)node * HIDF) + g;
    float4 v = *ip;
    v.x *= s; v.y *= s; v.z *= s; v.w *= s;
    *((float4*)(out + (size_t)node * HIDF) + g) = v;
}

// out[dst[e], :] += tmp[src[e], :]  (32 threads x float4 per edge, f32 atomics)
__global__ __launch_bounds__(256) void GNN_scatter(const int* __restrict__ src,
                                                   const int* __restrict__ dst,
                                                   const float* __restrict__ tmp,
                                                   float* __restrict__ out) {
    long t = (long)blockIdx.x * blockDim.x + threadIdx.x; // 0 .. NE*32-1
    int e = (int)(t >> 5);
    int g = (int)(t & 31);
    if (e >= NE) return;
    int s = src[e];
    int d = dst[e];
    float4 v = *((const float4*)(tmp + (size_t)s * HIDF) + g);
    float* op = out + (size_t)d * HIDF + (size_t)g * 4;
    atomicAdd(op + 0, v.x);
    atomicAdd(op + 1, v.y);
    atomicAdd(op + 2, v.z);
    atomicAdd(op + 3, v.w);
}

// -------------------------------------------------- TAGConv fused GEMM (WMMA)
// out[N,128] = relu( [h | a1 | a2] (N x 384)  @  W (384 x 128)  + b )
// One block = 16 rows; 8 waves each own a 16(M) x 16(N) tile.
// fp32 WMMA (V_WMMA_F32_16X16X4_F32) keeps full reference precision;
// W is staged through LDS in 64x128-row chunks (32 KB).
__global__ __launch_bounds__(256) void GNN_tag_gemm(const float* __restrict__ h,
                                                    const float* __restrict__ a1,
                                                    const float* __restrict__ a2,
                                                    const float* __restrict__ W,
                                                    const float* __restrict__ bias,
                                                    float* __restrict__ out) {
    __shared__ float lW[64 * HIDF];   // 32 KB

    const int tid  = threadIdx.x;
    const int wave = tid >> 5;        // 0..7 -> N-tile
    const int lane = tid & 31;
    const int half = lane >> 4;       // 0: lanes 0-15, 1: lanes 16-31
    const int l15  = lane & 15;

    const int m0  = blockIdx.x * 16;  // 50000 / 16 = 3125 exact
    const int n0  = wave * 16;
    const int row = m0 + l15;         // A-matrix M index (both halves)
    const int col = n0 + l15;         // B/C/D N index

    v8f acc = {0.0f, 0.0f, 0.0f, 0.0f, 0.0f, 0.0f, 0.0f, 0.0f};

    for (int kc = 0; kc < 6; ++kc) {          // 6 chunks of 64 K-rows = 384
        __syncthreads();
        // stage W[kc*64 .. kc*64+64, :] -> LDS (8192 floats, 2048 float4)
        {
            const float4* gw  = (const float4*)(W + (size_t)kc * 64 * HIDF);
            float4*       lw4 = (float4*)lW;
            #pragma unroll
            for (int i = 0; i < (64 * HIDF / 4) / 256; ++i)
                lw4[tid + i * 256] = gw[tid + i * 256];
        }
        __syncthreads();

        // which slice of the concatenated X does this K-chunk read?
        const int sel = kc >> 1;               // 0:h 1:a1 2:a2
        const float* xbase = (sel == 0) ? h : ((sel == 1) ? a1 : a2);
        const float* xp = xbase + (size_t)row * HIDF + (kc & 1) * 64;

        #pragma unroll
        for (int ks = 0; ks < 16; ++ks) {      // 16 x K=4 steps per chunk
            const int kl = ks * 4 + half * 2;  // local K: lanes0-15 K=0,1 / 16-31 K=2,3
            // A fragment: 16x4 f32 (ISA 7.12.2 layout)
            v2f a;
            {
                float2 av = *(const float2*)(xp + kl);
                a.x = av.x; a.y = av.y;
            }
            // B fragment: 4x16 f32 from LDS
            v2f b;
            b.x = lW[(kl + 0) * HIDF + col];
            b.y = lW[(kl + 1) * HIDF + col];
            acc = __builtin_amdgcn_wmma_f32_16x16x4_f32(
                /*neg_a=*/false, a, /*neg_b=*/false, b,
                /*c_mod=*/(short)0, acc, /*reuse_a=*/false, /*reuse_b=*/false);
        }
    }

    // D layout: VGPR v -> M = m0 + v + half*8, N = col. Fused bias + ReLU.
    const float bc = bias[col];
    #pragma unroll
    for (int v = 0; v < 8; ++v) {
        const int m = m0 + v + half * 8;
        float r = acc[v] + bc;
        out[(size_t)m * HIDF + col] = (r > 0.0f) ? r : 0.0f;
    }
}

// ------------------------------------------------------------- mean pooling

// hg[col] += sum over assigned rows of h[row, col]   (hg pre-zeroed)
__global__ __launch_bounds__(128) void GNN_pool(const float* __restrict__ h,
                                                float* __restrict__ hg) {
    const int colv = threadIdx.x;     // 0..127
    float s = 0.0f;
    for (int row = blockIdx.x; row < NN; row += gridDim.x)
        s += h[(size_t)row * HIDF + colv];
    atomicAdd(&hg[colv], s);
}

// out[0] = (hg/N) . Wp + bp
__global__ __launch_bounds__(128) void GNN_final(const float* __restrict__ hg,
                                                 const float* __restrict__ Wp,
                                                 const float* __restrict__ bp,
                                                 float* __restrict__ out) {
    __shared__ float red[128];
    const int t = threadIdx.x;
    red[t] = hg[t] * (1.0f / (float)NN) * Wp[t];
    __syncthreads();
    #pragma unroll
    for (int s = 64; s > 0; s >>= 1) {
        if (t < s) red[t] += red[t + s];
        __syncthreads();
    }
    if (t == 0) out[0] = red[0] + bp[0];
}

// ------------------------------------------------------------------- driver

extern "C" void kernel_launch(void* const* d_in, const int* in_sizes, int n_in,
                              void* d_out, int out_size, void* d_ws, size_t ws_size,
                              hipStream_t stream) {
    (void)in_sizes; (void)n_in; (void)out_size; (void)ws_size;

    const float* x   = (const float*)d_in[0];
    const int*   src = (const int*)  d_in[1];
    const int*   dst = (const int*)  d_in[2];
    const float* Wl[5] = { (const float*)d_in[3], (const float*)d_in[5],
                           (const float*)d_in[7], (const float*)d_in[9],
                           (const float*)d_in[11] };
    const float* bl[5] = { (const float*)d_in[4], (const float*)d_in[6],
                           (const float*)d_in[8], (const float*)d_in[10],
                           (const float*)d_in[12] };
    const float* Wp = (const float*)d_in[13];
    const float* bp = (const float*)d_in[14];
    float* outp = (float*)d_out;

    // workspace carve-up
    const size_t NF = (size_t)NN * HIDF;      // 6.4M floats per feature buffer
    float* base = (float*)d_ws;
    size_t o = 0;
    float* norm = base + o; o += 50176;       // deg -> norm in place (64-aligned)
    float* hg   = base + o; o += 128;
    float* hA   = base + o; o += NF;
    float* hB   = base + o; o += NF;
    float* a1   = base + o; o += NF;
    float* a2   = base + o; o += NF;
    float* tmp  = base + o; o += NF;
    float* sbuf = base + o; o += NF;

    const int nfBlocks   = (int)((NF + 255) / 256);      // per-element over N*128
    const int scaleBlks  = (NN * 32) / 256;              // 6250, exact
    const int scatBlks   = (NE * 32) / 256;              // 100000, exact
    const int degBlks    = NE / 256;                     // 3125, exact
    const int nBlks      = (NN + 255) / 256;             // 196
    const int gemmBlks   = NN / 16;                      // 3125, exact

    // degree + symmetric norm
    GNN_zero_f32<<<nBlks, 256, 0, stream>>>(norm, (long)NN);
    GNN_degree<<<degBlks, 256, 0, stream>>>(dst, norm);
    GNN_norm<<<nBlks, 256, 0, stream>>>(norm);

    // h = nan_to_num(x)
    GNN_nan_to_num<<<nfBlocks, 256, 0, stream>>>(x, hA, (long)NF);

    float* hcur  = hA;
    float* hnext = hB;
    for (int L = 0; L < 5; ++L) {
        // hop 1: a1 = norm * A * (norm * h)
        GNN_scale<<<scaleBlks, 256, 0, stream>>>(hcur, norm, tmp);
        GNN_zero_f32<<<nfBlocks, 256, 0, stream>>>(sbuf, (long)NF);
        GNN_scatter<<<scatBlks, 256, 0, stream>>>(src, dst, tmp, sbuf);
        GNN_scale<<<scaleBlks, 256, 0, stream>>>(sbuf, norm, a1);
        // hop 2: a2 = norm * A * (norm * a1)
        GNN_scale<<<scaleBlks, 256, 0, stream>>>(a1, norm, tmp);
        GNN_zero_f32<<<nfBlocks, 256, 0, stream>>>(sbuf, (long)NF);
        GNN_scatter<<<scatBlks, 256, 0, stream>>>(src, dst, tmp, sbuf);
        GNN_scale<<<scaleBlks, 256, 0, stream>>>(sbuf, norm, a2);
        // fused concat-GEMM + bias + ReLU (fp32 WMMA)
        GNN_tag_gemm<<<gemmBlks, 256, 0, stream>>>(hcur, a1, a2, Wl[L], bl[L], hnext);
        float* t2 = hcur; hcur = hnext; hnext = t2;
    }

    // mean over nodes, then final projection
    GNN_zero_f32<<<1, 128, 0, stream>>>(hg, 128L);
    GNN_pool<<<256, 128, 0, stream>>>(hcur, hg);
    GNN_final<<<1, 128, 0, stream>>>(hg, Wp, bp, outp);
}